// OVDClassifier_32014686225101
// MI455X (gfx1250) — compile-verified
//
#include <hip/hip_runtime.h>
#include <math.h>

// ---------------------------------------------------------------------------
// OVD box-similarity, reformulated:
//   out[b,m,n] = (1/area) * sum_d emb[n,d] * sum_k (u_m[h_k]*v_m[w_k]) * feats[b,k,d]
// where u/v come from column-cumsums of the 560x80 bicubic matrix.
// Dominant work: fp32 WMMA GEMM  g = omega x feats   (padded 128 rows/batch)
// then           out = g x emb^T.
// Each GEMM-g wave computes 4 m-tiles x 1 d-tile: B fragment loaded once,
// fed to 4 v_wmma_f32_16x16x4_f32 with 4 independent accumulators.
// ---------------------------------------------------------------------------

typedef float v2f __attribute__((ext_vector_type(2)));
typedef float v8f __attribute__((ext_vector_type(8)));

#define PGRID  80
#define TARGET 560
#define BATCH  2
#define MBOX   100
#define MPAD   128          // 8 tiles of 16, zero-padded rows 100..127
#define KTOT   6400
#define DDIM   1024
#define NEMB   128

// workspace layout (float offsets)
#define OFF_CM 0                                   // [561][80] column cumsums of Mh
#define OFF_MD (OFF_CM + (TARGET + 1) * PGRID)     // [560][80] dense bicubic matrix
#define OFF_OM (OFF_MD + TARGET * PGRID)           // [2*128][6400] per-box weights
#define OFF_G  (OFF_OM + BATCH * MPAD * KTOT)      // [2*128][1024] intermediate g

// ---------------------------------------------------------------------------
// Kernel 1: dense bicubic matrix (a = -0.75, align_corners=False, edge clamp)
// then per-column prefix sums CM[y][h] = sum_{H<y} M[H][h].  One block.
// ---------------------------------------------------------------------------
__global__ void __launch_bounds__(1024) k_bicubic_cum(float* ws) {
    float* Md = ws + OFF_MD;
    float* CM = ws + OFF_CM;
    const int t = threadIdx.x;

    for (int i = t; i < TARGET * PGRID; i += blockDim.x) Md[i] = 0.f;
    __syncthreads();

    if (t < TARGET) {
        const float a = -0.75f;
        float src = (t + 0.5f) * ((float)PGRID / (float)TARGET) - 0.5f;
        float ff  = floorf(src);
        int   f   = (int)ff;
        float tt  = src - ff;
        #pragma unroll
        for (int k = -1; k <= 2; ++k) {
            float x = fabsf(tt - (float)k);
            float w;
            if (x <= 1.f)      w = ((a + 2.f) * x - (a + 3.f)) * x * x + 1.f;
            else if (x < 2.f)  w = ((a * x - 5.f * a) * x + 8.f * a) * x - 4.f * a;
            else               w = 0.f;
            int idx = f + k;
            idx = idx < 0 ? 0 : (idx > PGRID - 1 ? PGRID - 1 : idx);
            Md[t * PGRID + idx] += w;     // row-private, same thread: safe
        }
    }
    __syncthreads();

    if (t < PGRID) {
        float run = 0.f;
        CM[t] = 0.f;
        for (int y = 1; y <= TARGET; ++y) {
            run += Md[(y - 1) * PGRID + t];
            CM[y * PGRID + t] = run;
        }
    }
}

// ---------------------------------------------------------------------------
// Kernel 2: per-box separable weights  omega[row][k] = u[k/80]*v[k%80]/area
// Rows 100..127 of each batch are zero padding. One block per padded row.
// ---------------------------------------------------------------------------
__global__ void __launch_bounds__(160) k_omega(const int* __restrict__ boxes,
                                               float* __restrict__ ws) {
    __shared__ float u[PGRID], v[PGRID];
    const float* CM = ws + OFF_CM;
    float* row = ws + OFF_OM + (size_t)blockIdx.x * KTOT;
    const int b = blockIdx.x / MPAD;
    const int m = blockIdx.x % MPAD;
    const int t = threadIdx.x;

    if (m >= MBOX) {                       // zero padding rows
        for (int k = t; k < KTOT; k += blockDim.x) row[k] = 0.f;
        return;
    }

    const int* bx = boxes + ((size_t)b * MBOX + m) * 4;
    const int x1 = bx[0], y1 = bx[1], w = bx[2], h = bx[3];
    const int xe = x1 + w - 1, ye = y1 + h - 1;
    const float inv_area = 1.f / ((float)(ye - y1) * (float)(xe - x1));

    if (t < PGRID)
        u[t] = (CM[ye * PGRID + t] - CM[y1 * PGRID + t]) * inv_area;
    else if (t < 2 * PGRID)
        v[t - PGRID] = CM[xe * PGRID + (t - PGRID)] - CM[x1 * PGRID + (t - PGRID)];
    __syncthreads();

    for (int k = t; k < KTOT; k += blockDim.x)
        row[k] = u[k / PGRID] * v[k % PGRID];
}

// ---------------------------------------------------------------------------
// Kernel 3: g = omega x feats via V_WMMA_F32_16X16X4_F32.
// One wave per (b, m-group of 4 tiles, d-tile) = 2*2*64 = 256 waves.
// Per K-step: 1 shared B fragment (2 b32 loads) -> 4 WMMAs with 4 A fragments
// (4 b64 loads) into 4 independent accumulators.  K loop = 6400/4.
// ---------------------------------------------------------------------------
__global__ void __launch_bounds__(256) k_gemm_g(const float* __restrict__ feats,
                                                float* __restrict__ ws) {
    const float* om = ws + OFF_OM;
    float*       g  = ws + OFF_G;
    const int wave = blockIdx.x * (blockDim.x >> 5) + (threadIdx.x >> 5);
    const int lane = threadIdx.x & 31;
    const int half = lane >> 4, l16 = lane & 15;

    const int dt  = wave & 63;
    const int tmp = wave >> 6;          // 0..3
    const int mg  = tmp & 1;            // m-group: tiles 4*mg .. 4*mg+3
    const int b   = tmp >> 1;

    const float* a0 = om + ((size_t)(b * MPAD) + (mg * 4 + 0) * 16 + l16) * KTOT + 2 * half;
    const float* a1 = om + ((size_t)(b * MPAD) + (mg * 4 + 1) * 16 + l16) * KTOT + 2 * half;
    const float* a2 = om + ((size_t)(b * MPAD) + (mg * 4 + 2) * 16 + l16) * KTOT + 2 * half;
    const float* a3 = om + ((size_t)(b * MPAD) + (mg * 4 + 3) * 16 + l16) * KTOT + 2 * half;
    const float* bcol = feats + (size_t)b * KTOT * DDIM
                              + (size_t)(2 * half) * DDIM + (dt * 16 + l16);

    v8f c0 = {}, c1 = {}, c2 = {}, c3 = {};
    #pragma unroll 4
    for (int k0 = 0; k0 < KTOT; k0 += 4) {
        v2f bb;
        bb.x = bcol[(size_t)k0 * DDIM];
        bb.y = bcol[(size_t)k0 * DDIM + DDIM];
        v2f x0 = *(const v2f*)(a0 + k0);
        v2f x1 = *(const v2f*)(a1 + k0);
        v2f x2 = *(const v2f*)(a2 + k0);
        v2f x3 = *(const v2f*)(a3 + k0);
        c0 = __builtin_amdgcn_wmma_f32_16x16x4_f32(false, x0, false, bb, (short)0, c0, false, false);
        c1 = __builtin_amdgcn_wmma_f32_16x16x4_f32(false, x1, false, bb, (short)0, c1, false, false);
        c2 = __builtin_amdgcn_wmma_f32_16x16x4_f32(false, x2, false, bb, (short)0, c2, false, false);
        c3 = __builtin_amdgcn_wmma_f32_16x16x4_f32(false, x3, false, bb, (short)0, c3, false, false);
    }
    // C/D layout: lane l, vgpr j -> (M = j + 8*(l/16), N = l%16)
    float* gout = g + ((size_t)(b * MPAD) + mg * 64 + 8 * half) * DDIM + dt * 16 + l16;
    #pragma unroll
    for (int j = 0; j < 8; ++j) {
        gout[(size_t)(j     ) * DDIM] = c0[j];
        gout[(size_t)(j + 16) * DDIM] = c1[j];
        gout[(size_t)(j + 32) * DDIM] = c2[j];
        gout[(size_t)(j + 48) * DDIM] = c3[j];
    }
}

// ---------------------------------------------------------------------------
// Kernel 4: out = g x emb^T via V_WMMA_F32_16X16X4_F32.
// One wave per (b, m-tile, n-tile) = 2*8*8 = 128 waves. K loop = 1024/4.
// ---------------------------------------------------------------------------
__global__ void __launch_bounds__(256) k_gemm_out(const float* __restrict__ emb,
                                                  float* __restrict__ out,
                                                  const float* __restrict__ ws) {
    const float* g = ws + OFF_G;
    const int wave = blockIdx.x * (blockDim.x >> 5) + (threadIdx.x >> 5);
    const int lane = threadIdx.x & 31;
    const int half = lane >> 4, l16 = lane & 15;

    const int nt  = wave & 7;
    const int tmp = wave >> 3;          // 0..15
    const int mt  = tmp & 7;
    const int b   = tmp >> 3;

    const float* arow = g   + ((size_t)(b * MPAD) + mt * 16 + l16) * DDIM + 2 * half;
    const float* brow = emb + (size_t)(nt * 16 + l16) * DDIM + 2 * half;

    v8f c = {};
    #pragma unroll 4
    for (int k0 = 0; k0 < DDIM; k0 += 4) {
        v2f a  = *(const v2f*)(arow + k0);
        v2f bb = *(const v2f*)(brow + k0);
        c = __builtin_amdgcn_wmma_f32_16x16x4_f32(
                false, a, false, bb, (short)0, c, false, false);
    }
    #pragma unroll
    for (int j = 0; j < 8; ++j) {
        int m = mt * 16 + j + 8 * half;
        if (m < MBOX)
            out[((size_t)(b * MBOX) + m) * NEMB + nt * 16 + l16] = c[j];
    }
}

// ---------------------------------------------------------------------------
extern "C" void kernel_launch(void* const* d_in, const int* in_sizes, int n_in,
                              void* d_out, int out_size, void* d_ws, size_t ws_size,
                              hipStream_t stream) {
    const float* feats = (const float*)d_in[0];   // [2,6400,1024] f32
    const float* emb   = (const float*)d_in[1];   // [128,1024]    f32
    const int*   boxes = (const int*)  d_in[2];   // [2,100,4]     i32
    float*       out   = (float*)d_out;           // [200,128]     f32
    float*       ws    = (float*)d_ws;            // ~8 MB used

    k_bicubic_cum<<<1, 1024, 0, stream>>>(ws);
    k_omega<<<BATCH * MPAD, 160, 0, stream>>>(boxes, ws);
    k_gemm_g<<<(BATCH * 2 * 64) / 8, 256, 0, stream>>>(feats, ws);     // 256 waves
    k_gemm_out<<<(BATCH * 8 * 8) / 8, 256, 0, stream>>>(emb, out, ws); // 128 waves
}